// UserItemGcn_24747601559683
// MI455X (gfx1250) — compile-verified
//
#include <hip/hip_runtime.h>
#include <hip/hip_bf16.h>

typedef __attribute__((ext_vector_type(2))) float v2f;
typedef __attribute__((ext_vector_type(8))) float v8f;

#define EMB_DIM 64

// ---------------------------------------------------------------------------
// float4 copy (seed d_out with the base embeddings)
// ---------------------------------------------------------------------------
__global__ void copy_f4_kernel(const float4* __restrict__ src,
                               float4* __restrict__ dst, int n4) {
    int t = blockIdx.x * blockDim.x + threadIdx.x;
    if (t < n4) dst[t] = src[t];
}

// ---------------------------------------------------------------------------
// float4 zero (clear scatter destination each hop)
// ---------------------------------------------------------------------------
__global__ void zero_f4_kernel(float4* __restrict__ dst, int n4) {
    int t = blockIdx.x * blockDim.x + threadIdx.x;
    if (t < n4) dst[t] = make_float4(0.f, 0.f, 0.f, 0.f);
}

// ---------------------------------------------------------------------------
// Edge-parallel scatter-add: dst[dIdx[e]] += src[sIdx[e]]  (rows of 64 f32)
// 16 lanes per edge, float4 gather (global_load_b128) + 4 f32 atomic adds.
// Both tables (38.4MB) fit in the 192MB L2, so gathers+atomics resolve at L2.
// ---------------------------------------------------------------------------
__global__ void scatter_add_kernel(const float* __restrict__ src,
                                   const int* __restrict__ sIdx,
                                   const int* __restrict__ dIdx,
                                   float* __restrict__ dst, int nEdges) {
    int t = blockIdx.x * blockDim.x + threadIdx.x;
    int e = t >> 4;
    int q = t & 15;
    if (e >= nEdges) return;
    int si = sIdx[e];
    int di = dIdx[e];
    const float4 v =
        *reinterpret_cast<const float4*>(src + (size_t)si * EMB_DIM + q * 4);
    float* d = dst + (size_t)di * EMB_DIM + q * 4;
    __hip_atomic_fetch_add(d + 0, v.x, __ATOMIC_RELAXED, __HIP_MEMORY_SCOPE_AGENT);
    __hip_atomic_fetch_add(d + 1, v.y, __ATOMIC_RELAXED, __HIP_MEMORY_SCOPE_AGENT);
    __hip_atomic_fetch_add(d + 2, v.z, __ATOMIC_RELAXED, __HIP_MEMORY_SCOPE_AGENT);
    __hip_atomic_fetch_add(d + 3, v.w, __ATOMIC_RELAXED, __HIP_MEMORY_SCOPE_AGENT);
}

__device__ inline float v8f_extract(v8f c, int i) {
    switch (i & 7) {
        case 0: return c[0]; case 1: return c[1];
        case 2: return c[2]; case 3: return c[3];
        case 4: return c[4]; case 5: return c[5];
        case 6: return c[6]; default: return c[7];
    }
}

// ---------------------------------------------------------------------------
// L2-normalize rows in place and accumulate into ui (lives in d_out).
// One wave32 handles 16 rows. Row-norms via V_WMMA_F32_16X16X4_F32:
// accumulate G = X_chunk * X_chunk^T over sixteen K=4 chunks; diag(G) is the
// squared row norms. A-layout (16x4 f32): lanes 0-15 hold K=0/1 in VGPR0/1,
// lanes 16-31 hold K=2/3 — identical register image for B = X^T, so A==B.
// C/D layout: diag row M is at lane M, vgpr M (M<8) / lane M+16, vgpr M-8.
// All 16 chunk loads are issued up-front into distinct registers so the
// compiler can clause them and drain the WMMA chain with partial waitcnts
// (round-1 version reloaded one register pair -> fully serialized chain).
// nRows is a multiple of 16, guard is wave-uniform => EXEC all-ones for WMMA.
// ---------------------------------------------------------------------------
__global__ void norm_accum_kernel(float* __restrict__ agg,
                                  float* __restrict__ ui, int nRows) {
    const int wavesPerBlock = blockDim.x >> 5;
    const int wave = blockIdx.x * wavesPerBlock + (threadIdx.x >> 5);
    const int lane = threadIdx.x & 31;
    const int row0 = wave * 16;
    if (row0 >= nRows) return;  // wave-uniform exit

    const int r = lane & 15;            // row within the 16-row tile
    const int khalf = (lane >> 4) << 1; // +0 (lanes 0-15) or +2 (lanes 16-31)
    const float* rowp = agg + (size_t)(row0 + r) * EMB_DIM;

    // issue all 16 independent chunk loads first (one s_clause, latency hidden)
    v2f a[16];
#pragma unroll
    for (int k = 0; k < 16; ++k)
        a[k] = *reinterpret_cast<const v2f*>(rowp + k * 4 + khalf);

    v8f c = {0.f, 0.f, 0.f, 0.f, 0.f, 0.f, 0.f, 0.f};
#pragma unroll
    for (int k = 0; k < 16; ++k) {
        // D = A * A^T + C : Gram matrix accumulation (8-arg VOP3P form)
        c = __builtin_amdgcn_wmma_f32_16x16x4_f32(
            /*neg_a=*/false, a[k], /*neg_b=*/false, a[k],
            /*c_mod=*/(short)0, c, /*reuse_a=*/false, /*reuse_b=*/false);
    }

    // Extract diagonal: lanes 0-7 -> rows 0-7 (c[lane]),
    //                   lanes 24-31 -> rows 8-15 (c[lane-24]).
    float nsq = (lane < 8) ? v8f_extract(c, lane) : v8f_extract(c, lane - 24);

    // Scaling pass: lane pair (2r, 2r+1) owns row r (cols 0-31 / 32-63).
    const int lr = lane >> 1;
    const int srcLane = (lr < 8) ? lr : (lr + 16);
    const float n2 = __shfl(nsq, srcLane, 32);
    const float scale = 1.0f / fmaxf(sqrtf(n2), 1e-12f);  // x / max(||x||, eps)

    const size_t rbase = (size_t)(row0 + lr) * EMB_DIM + (size_t)(lane & 1) * 32;
    float4* aggp = reinterpret_cast<float4*>(agg + rbase);
    float4* uip = reinterpret_cast<float4*>(ui + rbase);
#pragma unroll
    for (int j = 0; j < 8; ++j) {
        float4 v = aggp[j];
        v.x *= scale; v.y *= scale; v.z *= scale; v.w *= scale;
        aggp[j] = v;  // normalized in place -> next hop's gather source
        float4 u = uip[j];
        u.x += v.x; u.y += v.y; u.z += v.z; u.w += v.w;
        uip[j] = u;
    }
}

// ---------------------------------------------------------------------------
// Host-side orchestration
// ---------------------------------------------------------------------------
extern "C" void kernel_launch(void* const* d_in, const int* in_sizes, int n_in,
                              void* d_out, int out_size, void* d_ws, size_t ws_size,
                              hipStream_t stream) {
    const float* user_emb = (const float*)d_in[0];
    const float* item_emb = (const float*)d_in[1];
    const int* user_idx = (const int*)d_in[2];
    const int* item_idx = (const int*)d_in[3];

    const int nU = in_sizes[0] / EMB_DIM;  // 100000
    const int nI = in_sizes[1] / EMB_DIM;  // 50000
    const int nE = in_sizes[2];            // 1000000

    const size_t uElems = (size_t)nU * EMB_DIM;
    const size_t iElems = (size_t)nI * EMB_DIM;

    // d_out = [item_ui (nI*64), user_ui (nU*64)], seeded with base embeddings
    float* outItem = (float*)d_out;
    float* outUser = outItem + iElems;

    // workspace: ping-pong aggregation buffers
    float* userA = (float*)d_ws;
    float* userB = userA + uElems;
    float* itemA = userB + uElems;
    float* itemB = itemA + iElems;

    const int BLK = 256;
    const int u4 = (int)(uElems / 4), i4 = (int)(iElems / 4);

    // seed ui accumulators with the embeddings (reference adds emb once)
    copy_f4_kernel<<<(i4 + BLK - 1) / BLK, BLK, 0, stream>>>(
        (const float4*)item_emb, (float4*)outItem, i4);
    copy_f4_kernel<<<(u4 + BLK - 1) / BLK, BLK, 0, stream>>>(
        (const float4*)user_emb, (float4*)outUser, u4);

    const float* uSrc = user_emb;
    const float* iSrc = item_emb;
    float* uDst = userA; float* uAlt = userB;
    float* iDst = itemA; float* iAlt = itemB;

    const int scatterBlocks = (nE * 16 + BLK - 1) / BLK;
    const int uWaves = nU / 16, iWaves = nI / 16;
    const int uNormBlocks = (uWaves * 32 + BLK - 1) / BLK;
    const int iNormBlocks = (iWaves * 32 + BLK - 1) / BLK;

    for (int hop = 0; hop < 3; ++hop) {
        zero_f4_kernel<<<(u4 + BLK - 1) / BLK, BLK, 0, stream>>>((float4*)uDst, u4);
        zero_f4_kernel<<<(i4 + BLK - 1) / BLK, BLK, 0, stream>>>((float4*)iDst, i4);

        // user_agg[u] += item_agg[i] over edges (u,i); and vice versa
        scatter_add_kernel<<<scatterBlocks, BLK, 0, stream>>>(
            iSrc, item_idx, user_idx, uDst, nE);
        scatter_add_kernel<<<scatterBlocks, BLK, 0, stream>>>(
            uSrc, user_idx, item_idx, iDst, nE);

        norm_accum_kernel<<<uNormBlocks, BLK, 0, stream>>>(uDst, outUser, nU);
        norm_accum_kernel<<<iNormBlocks, BLK, 0, stream>>>(iDst, outItem, nI);

        uSrc = uDst; iSrc = iDst;
        float* t;
        t = uDst; uDst = uAlt; uAlt = t;
        t = iDst; iDst = iAlt; iAlt = t;
    }
}